// CLUB_14061722927686
// MI455X (gfx1250) — compile-verified
//
#include <hip/hip_runtime.h>
#include <math.h>

typedef __attribute__((ext_vector_type(2))) float v2f;
typedef __attribute__((ext_vector_type(8))) float v8f;

#define N_ROWS 1024   // batch
#define KDIM   256    // inner dim of every GEMM
#define NDIM   256    // output dim of every GEMM
#define X2D    128
#define BSTRIDE 260   // 16-col W slice row stride in LDS (pad: 4r bank rotation)

#if __has_builtin(__builtin_amdgcn_global_load_async_to_lds_b32)
#define HAVE_ASYNC_LDS 1
#else
#define HAVE_ASYNC_LDS 0
#endif

typedef __attribute__((address_space(1))) int* as1_i32p;
typedef __attribute__((address_space(3))) int* as3_i32p;
typedef __attribute__((address_space(1))) void* as1_vp;
typedef __attribute__((address_space(3))) void* as3_vp;

// One 32-bit global->LDS async copy (falls back to load+store if builtin absent)
__device__ __forceinline__ void copy_b32_to_lds(const float* gsrc, float* ldst) {
#if HAVE_ASYNC_LDS
    __builtin_amdgcn_global_load_async_to_lds_b32(
        (as1_i32p)(as1_vp)(float*)gsrc,   // addrspace(1) int*
        (as3_i32p)(as3_vp)ldst,           // addrspace(3) int*
        /*imm_offset=*/0, /*cpol=*/0);
#else
    *ldst = *gsrc;
#endif
}

// ---------------------------------------------------------------------------
// Column moments of x2: m1[d] = mean_j x2[j,d], m2[d] = mean_j x2[j,d]^2
// ---------------------------------------------------------------------------
__global__ __launch_bounds__(128) void x2_stats(const float* __restrict__ x2,
                                                float* __restrict__ m1,
                                                float* __restrict__ m2) {
    const int d = threadIdx.x;               // 128 threads = 128 columns
    float s1 = 0.0f, s2 = 0.0f;
    #pragma unroll 8
    for (int j = 0; j < N_ROWS; ++j) {
        float v = x2[j * X2D + d];
        s1 += v;
        s2 += v * v;
    }
    m1[d] = s1 * (1.0f / (float)N_ROWS);
    m2[d] = s2 * (1.0f / (float)N_ROWS);
}

// ---------------------------------------------------------------------------
// out[M,N] = act(A[M,K] @ W[K,N] + bias[N])   with M=1024, K=N=256
//
// Block = 8 waves = one 16-wide column slice x 128 rows of output.
// The block's W slice (256x16 = 16KB) is staged ONCE into LDS, transposed to
// [col][k] (row stride 260 floats -> conflict-free ds_load_b64 B fragments),
// via GLOBAL_LOAD_ASYNC_TO_LDS_B32 when available (ASYNCcnt path).
//
// Each wave then runs one 16x16 tile with V_WMMA_F32_16X16X4_F32, K-loop of 64
// issues; A fragments are contiguous global b64 loads.
//
// WMMA f32 16x16x4 fragment layout (ISA 7.12.2):
//   A: lanes 0-15 rows M=0..15, K={0,1}; lanes 16-31 K={2,3}
//   B: VGPR v, lanes 0-15: K=v, N=lane; lanes 16-31: K=v+2, N=lane-16
//   C/D: VGPR v, lanes 0-15: M=v, N=lane; lanes 16-31: M=v+8
// ACT: 0 = relu, 1 = tanh
// ---------------------------------------------------------------------------
template <int ACT>
__global__ __launch_bounds__(256) void gemm_bias_act(const float* __restrict__ A,
                                                     const float* __restrict__ W,
                                                     const float* __restrict__ bias,
                                                     float* __restrict__ out) {
    __shared__ __align__(16) float Bsm[16][BSTRIDE];

    const int lane  = threadIdx.x & 31;
    const int wave  = threadIdx.x >> 5;
    const int tileN = blockIdx.x & 15;           // 16 column groups
    const int mgrp  = blockIdx.x >> 4;           // 8 row groups of 128 rows
    const int tileM = mgrp * 8 + wave;
    const int r     = lane & 15;
    const int half  = lane >> 4;                 // selects K sub-pair

    // ---- Stage W[:, tileN*16 .. +15] into LDS, transposed to [col][k] ----
    const float* __restrict__ wsrc = W + tileN * 16;
    #pragma unroll
    for (int i = threadIdx.x; i < KDIM * 16; i += 256) {
        const int k = i >> 4;
        const int c = i & 15;
        copy_b32_to_lds(wsrc + (size_t)k * NDIM + c, &Bsm[c][k]);
    }
#if HAVE_ASYNC_LDS
#if __has_builtin(__builtin_amdgcn_s_wait_asynccnt)
    __builtin_amdgcn_s_wait_asynccnt(0);
#else
    asm volatile("s_wait_asynccnt 0x0" ::: "memory");
#endif
#endif
    __syncthreads();

    // ---- K loop: 64 x V_WMMA_F32_16X16X4_F32 ----
    const float* __restrict__ ap   = A + ((size_t)(tileM * 16 + r)) * KDIM + 2 * half;
    const float* __restrict__ brow = &Bsm[r][2 * half];

    v8f acc = {};
    #pragma unroll 8
    for (int k = 0; k < KDIM; k += 4) {
        v2f a = *(const v2f*)(ap + k);           // A[row, k+2h .. k+2h+1] (b64)
        v2f b = *(const v2f*)(brow + k);         // W[k+2h .. k+2h+1, col] (ds b64)
        acc = __builtin_amdgcn_wmma_f32_16x16x4_f32(
            /*neg_a=*/false, a, /*neg_b=*/false, b,
            /*c_mod=*/(short)0, acc, /*reuse_a=*/false, /*reuse_b=*/false);
    }

    // ---- Epilogue: bias + activation, store 8 rows per lane ----
    const int   n  = tileN * 16 + r;
    const float bn = bias[n];
    float* __restrict__ op = out + ((size_t)(tileM * 16 + 8 * half)) * NDIM + n;
    #pragma unroll
    for (int v = 0; v < 8; ++v) {
        float x = acc[v] + bn;
        if (ACT == 0) x = x > 0.0f ? x : 0.0f;
        else          x = tanhf(x);
        op[(size_t)v * NDIM] = x;
    }
}

// ---------------------------------------------------------------------------
// Per-row CLUB contribution using the moment trick:
//   contrib(i,d) = -0.5*iv*[(mu-x2)^2 - (mu^2 - 2*mu*m1 + m2)]
// One block per row, 128 threads (one per d), deterministic LDS tree reduce.
// ---------------------------------------------------------------------------
__global__ __launch_bounds__(128) void row_contrib(const float* __restrict__ g,
                                                   const float* __restrict__ x2,
                                                   const float* __restrict__ m1,
                                                   const float* __restrict__ m2,
                                                   float* __restrict__ rowsum) {
    __shared__ float sm[128];
    const int i = blockIdx.x;
    const int d = threadIdx.x;

    float mu  = g[(size_t)i * NDIM + d];
    float lv  = g[(size_t)i * NDIM + X2D + d];
    float iv  = expf(-lv);
    float xv  = x2[(size_t)i * X2D + d];
    float dm  = mu - xv;
    float pos = -0.5f * dm * dm * iv;
    float neg = -0.5f * (mu * mu - 2.0f * mu * m1[d] + m2[d]) * iv;
    sm[d] = pos - neg;
    __syncthreads();

    #pragma unroll
    for (int s = 64; s > 0; s >>= 1) {
        if (d < s) sm[d] += sm[d + s];
        __syncthreads();
    }
    if (d == 0) rowsum[i] = sm[0];
}

__global__ __launch_bounds__(256) void final_reduce(const float* __restrict__ rowsum,
                                                    float* __restrict__ out) {
    __shared__ float sm[256];
    float s = 0.0f;
    for (int i = threadIdx.x; i < N_ROWS; i += 256) s += rowsum[i];
    sm[threadIdx.x] = s;
    __syncthreads();
    #pragma unroll
    for (int st = 128; st > 0; st >>= 1) {
        if (threadIdx.x < st) sm[threadIdx.x] += sm[threadIdx.x + st];
        __syncthreads();
    }
    if (threadIdx.x == 0) out[0] = sm[0] * (1.0f / (float)N_ROWS);
}

// ---------------------------------------------------------------------------
extern "C" void kernel_launch(void* const* d_in, const int* in_sizes, int n_in,
                              void* d_out, int out_size, void* d_ws, size_t ws_size,
                              hipStream_t stream) {
    const float* x1 = (const float*)d_in[0];
    const float* x2 = (const float*)d_in[1];
    const float* W1 = (const float*)d_in[2];
    const float* b1 = (const float*)d_in[3];
    const float* W2 = (const float*)d_in[4];
    const float* b2 = (const float*)d_in[5];
    const float* W3 = (const float*)d_in[6];
    const float* b3 = (const float*)d_in[7];
    float* out = (float*)d_out;

    char* ws = (char*)d_ws;
    const size_t MAT = (size_t)N_ROWS * NDIM * sizeof(float);   // 1 MB
    float* h1     = (float*)(ws);
    float* h2     = (float*)(ws + MAT);
    float* g      = (float*)(ws + 2 * MAT);
    float* m1     = (float*)(ws + 3 * MAT);
    float* m2     = m1 + X2D;
    float* rowsum = m2 + X2D;

    x2_stats<<<1, 128, 0, stream>>>(x2, m1, m2);
    gemm_bias_act<0><<<128, 256, 0, stream>>>(x1, W1, b1, h1);
    gemm_bias_act<0><<<128, 256, 0, stream>>>(h1, W2, b2, h2);
    gemm_bias_act<1><<<128, 256, 0, stream>>>(h2, W3, b3, g);
    row_contrib<<<N_ROWS, 128, 0, stream>>>(g, x2, m1, m2, rowsum);
    final_reduce<<<1, 256, 0, stream>>>(rowsum, out);
}